// MambaEncoder_59863254172033
// MI455X (gfx1250) — compile-verified
//
#include <hip/hip_runtime.h>
#include <hip/hip_bf16.h>
#include <math.h>

// ---------------------------------------------------------------------------
// MambaEncoder for MI455X (gfx1250).  fp32 end-to-end; all dense projections
// run through V_WMMA_F32_16X16X4_F32 (full-precision matrix pipe).
// GEMM register-blocks N by 4: each wave computes a 16x64 strip, reusing the
// A-fragment across 4 independent WMMA accumulator chains per k-step.
// Selective scan parallelized over state channels w/ lane-shuffle reductions.
// Workspace requirement: ~96 MB (bump-allocated below).
// ---------------------------------------------------------------------------

typedef float v2f __attribute__((ext_vector_type(2)));
typedef float v8f __attribute__((ext_vector_type(8)));

enum { ACT_NONE = 0, ACT_GELU = 1, ACT_SILU = 2, ACT_SOFTPLUS = 3 };

__device__ __forceinline__ float silu_f(float x)  { return x / (1.0f + __expf(-x)); }
__device__ __forceinline__ float gelu_f(float x)  { return 0.5f * x * (1.0f + erff(x * 0.70710678118654752f)); }
__device__ __forceinline__ float softplus_f(float x) { return (x > 20.0f) ? x : log1pf(__expf(x)); }

#define WMMA_F32(ACC, Afrag, Bfrag)                                            \
    (ACC) = __builtin_amdgcn_wmma_f32_16x16x4_f32(false, (Afrag), false,       \
                                                  (Bfrag), (short)0, (ACC),    \
                                                  false, false)

// ---------------------------------------------------------------------------
// WMMA f32 GEMM:  C[M,N] = act(A[M,K] * B[K,N] + bias) + resid
//   A row-major, leading dim lda.  B row-major (K,N) or, if transB, (N,K)
//   with leading dim ldb.  One wave -> 16x64 strip (4 accumulators), 4 waves
//   per block -> 16x256 per block.
// A-frag layout (16x4 f32): v0 = {lanes0-15: K+0 | lanes16-31: K+2},
//                           v1 = {K+1 | K+3}, M = lane%16.
// B-frag layout (4x16 f32): mirrored, N = lane%16.
// C/D layout: VGPR r holds rows r (lanes0-15) and r+8 (lanes16-31).
// ---------------------------------------------------------------------------
__global__ __launch_bounds__(128)
void gemm_wmma_f32(const float* __restrict__ A, int lda,
                   const float* __restrict__ B, int ldb, int transB,
                   const float* __restrict__ bias,
                   const float* __restrict__ resid,
                   float* __restrict__ C,
                   int M, int N, int K, int act)
{
    const int wave = threadIdx.x >> 5;
    const int lane = threadIdx.x & 31;
    const int half = lane >> 4;
    const int l16  = lane & 15;
    const int m0    = blockIdx.y * 16;
    const int nbase = (blockIdx.x * 4 + wave) * 64;
    if (m0 >= M || nbase >= N) return;

    const int row  = m0 + l16;
    const int col0 = nbase + l16;
    v8f acc[4] = {};

    const bool interior = (m0 + 16 <= M) && (nbase + 64 <= N);
    int k = 0;
    if (interior) {
        const float* Ar = A + (size_t)row * lda;
        if (transB) {
            const float* Bc0 = B + (size_t)(col0     ) * ldb;
            const float* Bc1 = B + (size_t)(col0 + 16) * ldb;
            const float* Bc2 = B + (size_t)(col0 + 32) * ldb;
            const float* Bc3 = B + (size_t)(col0 + 48) * ldb;
            for (; k + 4 <= K; k += 4) {
                const int ka = k + 2 * half;
                v2f a  = { Ar[ka],  Ar[ka + 1]  };
                v2f b0 = { Bc0[ka], Bc0[ka + 1] };
                v2f b1 = { Bc1[ka], Bc1[ka + 1] };
                v2f b2 = { Bc2[ka], Bc2[ka + 1] };
                v2f b3 = { Bc3[ka], Bc3[ka + 1] };
                WMMA_F32(acc[0], a, b0);
                WMMA_F32(acc[1], a, b1);
                WMMA_F32(acc[2], a, b2);
                WMMA_F32(acc[3], a, b3);
            }
        } else {
            for (; k + 4 <= K; k += 4) {
                const int ka = k + 2 * half;
                v2f a = { Ar[ka], Ar[ka + 1] };
                const float* Br0 = B + (size_t)ka * ldb;
                const float* Br1 = Br0 + ldb;
                v2f b0 = { Br0[col0],      Br1[col0]      };
                v2f b1 = { Br0[col0 + 16], Br1[col0 + 16] };
                v2f b2 = { Br0[col0 + 32], Br1[col0 + 32] };
                v2f b3 = { Br0[col0 + 48], Br1[col0 + 48] };
                WMMA_F32(acc[0], a, b0);
                WMMA_F32(acc[1], a, b1);
                WMMA_F32(acc[2], a, b2);
                WMMA_F32(acc[3], a, b3);
            }
        }
    }
    // Guarded tail (K remainder, or the entire loop for edge strips).
    for (; k < K; k += 4) {
        const int ka = k + 2 * half;
        v2f a = {0.0f, 0.0f};
        if (row < M) {
            if (ka     < K) a.x = A[(size_t)row * lda + ka];
            if (ka + 1 < K) a.y = A[(size_t)row * lda + ka + 1];
        }
#pragma unroll
        for (int j = 0; j < 4; ++j) {
            const int col = col0 + 16 * j;
            v2f b = {0.0f, 0.0f};
            if (col < N) {
                if (transB) {
                    if (ka     < K) b.x = B[(size_t)col * ldb + ka];
                    if (ka + 1 < K) b.y = B[(size_t)col * ldb + ka + 1];
                } else {
                    if (ka     < K) b.x = B[(size_t)ka       * ldb + col];
                    if (ka + 1 < K) b.y = B[(size_t)(ka + 1) * ldb + col];
                }
            }
            WMMA_F32(acc[j], a, b);
        }
    }

    // Epilogue: bias -> activation -> residual -> store (per-tile col guard).
#pragma unroll
    for (int j = 0; j < 4; ++j) {
        const int col = col0 + 16 * j;
        if (col >= N) continue;
        const float bcol = (bias != nullptr) ? bias[col] : 0.0f;
#pragma unroll
        for (int r = 0; r < 8; ++r) {
            const int rm = m0 + r + 8 * half;
            if (rm < M) {
                float v = acc[j][r] + bcol;
                if      (act == ACT_GELU)     v = gelu_f(v);
                else if (act == ACT_SILU)     v = silu_f(v);
                else if (act == ACT_SOFTPLUS) v = softplus_f(v);
                if (resid) v += resid[(size_t)rm * N + col];
                C[(size_t)rm * N + col] = v;
            }
        }
    }
}

// ---------------------------------------------------------------------------
// LayerNorm over last dim C; one block per row.  Safe in-place.
// ---------------------------------------------------------------------------
__global__ __launch_bounds__(256)
void layernorm_k(const float* __restrict__ x, const float* __restrict__ g,
                 const float* __restrict__ b, float* __restrict__ out, int C)
{
    __shared__ float s1[256], s2[256];
    const size_t row = blockIdx.x;
    const float* xr = x + row * (size_t)C;
    float a = 0.0f, q = 0.0f;
    for (int c = threadIdx.x; c < C; c += 256) { float v = xr[c]; a += v; q += v * v; }
    s1[threadIdx.x] = a; s2[threadIdx.x] = q;
    __syncthreads();
    for (int off = 128; off > 0; off >>= 1) {
        if ((int)threadIdx.x < off) {
            s1[threadIdx.x] += s1[threadIdx.x + off];
            s2[threadIdx.x] += s2[threadIdx.x + off];
        }
        __syncthreads();
    }
    const float mean = s1[0] / C;
    const float var  = s2[0] / C - mean * mean;
    const float rstd = rsqrtf(var + 1e-5f);
    float* orow = out + row * (size_t)C;
    for (int c = threadIdx.x; c < C; c += 256)
        orow[c] = (xr[c] - mean) * rstd * g[c] + b[c];
}

// LayerNorm + write transposed (B,L,C) -> (B,C,L) stage output.
__global__ __launch_bounds__(256)
void ln_transpose_k(const float* __restrict__ x, const float* __restrict__ g,
                    const float* __restrict__ b, float* __restrict__ out, int L, int C)
{
    __shared__ float s1[256], s2[256];
    const size_t row = blockIdx.x;
    const float* xr = x + row * (size_t)C;
    float a = 0.0f, q = 0.0f;
    for (int c = threadIdx.x; c < C; c += 256) { float v = xr[c]; a += v; q += v * v; }
    s1[threadIdx.x] = a; s2[threadIdx.x] = q;
    __syncthreads();
    for (int off = 128; off > 0; off >>= 1) {
        if ((int)threadIdx.x < off) {
            s1[threadIdx.x] += s1[threadIdx.x + off];
            s2[threadIdx.x] += s2[threadIdx.x + off];
        }
        __syncthreads();
    }
    const float mean = s1[0] / C;
    const float var  = s2[0] / C - mean * mean;
    const float rstd = rsqrtf(var + 1e-5f);
    for (int c = threadIdx.x; c < C; c += 256)
        out[(size_t)c * L + row] = (xr[c] - mean) * rstd * g[c] + b[c];
}

// ---------------------------------------------------------------------------
// Patch-embed gather: (1,2,96,96,96) -> P[13824,128], j = i*64+kd*16+kh*4+kw
// (matches pe_w (O,I,4,4,4) flattening so the conv becomes a GEMM).
// ---------------------------------------------------------------------------
__global__ void patch_gather_k(const float* __restrict__ x, float* __restrict__ P)
{
    const int idx = blockIdx.x * blockDim.x + threadIdx.x;
    if (idx >= 13824 * 128) return;
    const int l = idx >> 7, j = idx & 127;
    const int i = j >> 6, r = j & 63;
    const int kd = r >> 4, kh = (r >> 2) & 3, kw = r & 3;
    const int d = l / 576, h = (l / 24) % 24, w = l % 24;
    P[idx] = x[(((size_t)i * 96 + 4 * d + kd) * 96 + 4 * h + kh) * 96 + 4 * w + kw];
}

// Causal depthwise conv1d (kernel 4) + SiLU.  Reads first `di` columns of the
// (L, stride) in_proj output.
__global__ void dwconv_silu_k(const float* __restrict__ xz, int stride,
                              const float* __restrict__ w, const float* __restrict__ bias,
                              float* __restrict__ xc, int L, int di)
{
    const int idx = blockIdx.x * blockDim.x + threadIdx.x;
    if (idx >= L * di) return;
    const int t = idx / di, d = idx % di;
    float acc = bias[d];
#pragma unroll
    for (int k = 0; k < 4; ++k) {
        const int tt = t - 3 + k;
        if (tt >= 0) acc += xz[(size_t)tt * stride + d] * w[d * 4 + k];
    }
    xc[idx] = silu_f(acc);
}

// ---------------------------------------------------------------------------
// Selective scan.  Each 16-lane half-wave owns channel d, lanes carry state
// index n=0..15.  h_t = exp(dt*A)*h + dt*B_t*xc ; y = <h,C_t> reduced across
// the 16 state lanes via lane shuffles, then gated with silu(z) and D skip.
// dA/dBu are recomputed on the fly (never materialized).
// ---------------------------------------------------------------------------
__global__ __launch_bounds__(256)
void selscan_k(const float* __restrict__ dt, const float* __restrict__ dbl, int ldd, int dtr,
               const float* __restrict__ xc, const float* __restrict__ A_log,
               const float* __restrict__ Dp, const float* __restrict__ xz, int ldxz,
               float* __restrict__ y, int L, int di)
{
    const int lane = threadIdx.x & 31;
    const int wv   = threadIdx.x >> 5;
    const int n    = lane & 15;
    const int d    = blockIdx.x * 16 + wv * 2 + (lane >> 4);
    if (d >= di) return;

    const float a  = -__expf(A_log[d * 16 + n]);
    const float Dd = Dp[d];
    float h = 0.0f;
    for (int t = 0; t < L; ++t) {
        const float dtt = dt[(size_t)t * di + d];
        const float xct = xc[(size_t)t * di + d];
        const float Bn  = dbl[(size_t)t * ldd + dtr + n];
        const float Cn  = dbl[(size_t)t * ldd + dtr + 16 + n];
        h = __expf(dtt * a) * h + dtt * Bn * xct;
        float p = h * Cn;
        p += __shfl_xor(p, 8, 32);
        p += __shfl_xor(p, 4, 32);
        p += __shfl_xor(p, 2, 32);
        p += __shfl_xor(p, 1, 32);
        if (n == 0) {
            const float z = xz[(size_t)t * ldxz + di + d];
            y[(size_t)t * di + d] = (p + xct * Dd) * silu_f(z);
        }
    }
}

// Patch-merge gather: (L=H*W*T, C) -> (L/8, 8C) with the reference's part order.
__global__ void merge_gather_k(const float* __restrict__ x, float* __restrict__ mrg,
                               int H, int W, int T, int C)
{
    const int H2 = H / 2, W2 = W / 2, T2 = T / 2;
    const size_t total = (size_t)H2 * W2 * T2 * 8 * C;
    const size_t idx = (size_t)blockIdx.x * blockDim.x + threadIdx.x;
    if (idx >= total) return;
    const int c = (int)(idx % C);
    size_t rest = idx / C;
    const int p = (int)(rest % 8);
    const size_t lp = rest / 8;
    const int t2 = (int)(lp % T2);
    const int w2 = (int)((lp / T2) % W2);
    const int h2 = (int)(lp / ((size_t)T2 * W2));
    const int dh[8] = {0, 1, 0, 0, 1, 0, 1, 1};
    const int dw[8] = {0, 0, 1, 0, 1, 1, 0, 1};
    const int dv[8] = {0, 0, 0, 1, 0, 1, 1, 1};
    const int hh = 2 * h2 + dh[p], ww = 2 * w2 + dw[p], tt = 2 * t2 + dv[p];
    mrg[idx] = x[(((size_t)hh * W + ww) * T + tt) * (size_t)C + c];
}

// ---------------------------------------------------------------------------
// Host orchestration
// ---------------------------------------------------------------------------
struct MambaP { const float *in_w, *conv_w, *conv_b, *x_w, *dt_w, *dt_b, *A_log, *D, *out_w; };
struct BlockP { const float *n1g, *n1b, *n2g, *n2b, *fc1_w, *fc1_b, *fc2_w, *fc2_b; MambaP m; };
struct DownP  { const float *ng, *nb, *red_w; };

extern "C" void kernel_launch(void* const* d_in, const int* in_sizes, int n_in,
                              void* d_out, int out_size, void* d_ws, size_t ws_size,
                              hipStream_t stream)
{
    (void)in_sizes; (void)n_in; (void)out_size; (void)ws_size;
    static const int depths[4] = {2, 2, 4, 2};

    // --- unpack inputs in setup_inputs() dict-insertion order (recursive) ---
    int ii = 0;
    auto F = [&](void) { return (const float*)d_in[ii++]; };
    const float* xin   = F();           // x (1,2,96,96,96)
    const float* pe_w  = F();           // (96, 2*4*4*4) after flatten
    const float* pe_b  = F();
    const float* pe_ng = F();
    const float* pe_nb = F();
    BlockP blocks[4][4];
    DownP  down[3];
    for (int s = 0; s < 4; ++s) {
        for (int bk = 0; bk < depths[s]; ++bk) {
            BlockP& B = blocks[s][bk];
            B.n1g = F(); B.n1b = F(); B.n2g = F(); B.n2b = F();
            MambaP& m = B.m;
            m.in_w = F(); m.conv_w = F(); m.conv_b = F(); m.x_w = F();
            m.dt_w = F(); m.dt_b = F(); m.A_log = F(); m.D = F(); m.out_w = F();
            B.fc1_w = F(); B.fc1_b = F(); B.fc2_w = F(); B.fc2_b = F();
        }
        if (s < 3) { down[s].ng = F(); down[s].nb = F(); down[s].red_w = F(); }
    }
    const float *on_g[4], *on_b[4];
    for (int s = 0; s < 4; ++s) { on_g[s] = F(); on_b[s] = F(); }

    // --- workspace bump allocator (floats) ---
    size_t off = 0;
    auto alloc = [&](size_t n) { float* p = (float*)d_ws + off; off += n; return p; };
    float* P_X   = alloc(1400000);   // tokens   (L*C)          max 1,327,104
    float* P_TMP = alloc(1400000);   // normed x
    float* P_XZ  = alloc(5400000);   // in_proj out / patch gather
    float* P_XC  = alloc(2700000);   // conv+silu out
    float* P_DBL = alloc(600000);    // x_proj out (L, dtr+32)
    float* P_DT  = alloc(2700000);   // softplus(dt)
    float* P_Y   = alloc(2700000);   // scan output
    float* P_HID = alloc(5400000);   // MLP hidden
    float* P_MRG = alloc(1400000);   // patch-merge concat (L/8, 8C)

    auto gemm = [&](const float* A, int lda, const float* B, int ldb, int tB,
                    const float* bias, const float* resid, float* C,
                    int M, int N, int K, int act) {
        dim3 grid((N + 255) / 256, (M + 15) / 16);
        gemm_wmma_f32<<<grid, 128, 0, stream>>>(A, lda, B, ldb, tB, bias, resid, C, M, N, K, act);
    };
    auto LN = [&](const float* x, const float* g, const float* b, float* o, int L, int C) {
        layernorm_k<<<L, 256, 0, stream>>>(x, g, b, o, C);
    };

    // --- patch embed: gather -> GEMM (pe_w is (N=96,K=128) row-major => transB) ---
    patch_gather_k<<<(13824 * 128 + 255) / 256, 256, 0, stream>>>(xin, P_XZ);
    gemm(P_XZ, 128, pe_w, 128, /*transB=*/1, pe_b, nullptr, P_X, 13824, 96, 128, ACT_NONE);
    LN(P_X, pe_ng, pe_nb, P_X, 13824, 96);   // in-place is safe

    int H = 24, W = 24, T = 24, L = 13824, dim = 96;
    float* outp = (float*)d_out;
    size_t out_off = 0;

    for (int s = 0; s < 4; ++s) {
        const int di  = 2 * dim;
        const int dtr = (dim + 15) / 16;
        const int hid = 4 * dim;
        const int wB  = dtr + 32;             // x_proj output width
        for (int bk = 0; bk < depths[s]; ++bk) {
            const BlockP& B = blocks[s][bk];
            // ---- Mamba branch ----
            LN(P_X, B.n1g, B.n1b, P_TMP, L, dim);
            gemm(P_TMP, dim, B.m.in_w, 2 * di, 0, nullptr, nullptr, P_XZ, L, 2 * di, dim, ACT_NONE);
            dwconv_silu_k<<<(L * di + 255) / 256, 256, 0, stream>>>(P_XZ, 2 * di, B.m.conv_w,
                                                                    B.m.conv_b, P_XC, L, di);
            gemm(P_XC, di, B.m.x_w, wB, 0, nullptr, nullptr, P_DBL, L, wB, di, ACT_NONE);
            gemm(P_DBL, wB, B.m.dt_w, di, 0, B.m.dt_b, nullptr, P_DT, L, di, dtr, ACT_SOFTPLUS);
            selscan_k<<<di / 16, 256, 0, stream>>>(P_DT, P_DBL, wB, dtr, P_XC, B.m.A_log,
                                                   B.m.D, P_XZ, 2 * di, P_Y, L, di);
            gemm(P_Y, di, B.m.out_w, dim, 0, nullptr, /*resid=*/P_X, P_X, L, dim, di, ACT_NONE);
            // ---- MLP branch ----
            LN(P_X, B.n2g, B.n2b, P_TMP, L, dim);
            gemm(P_TMP, dim, B.fc1_w, hid, 0, B.fc1_b, nullptr, P_HID, L, hid, dim, ACT_GELU);
            gemm(P_HID, hid, B.fc2_w, dim, 0, B.fc2_b, /*resid=*/P_X, P_X, L, dim, hid, ACT_NONE);
        }
        // ---- stage output: outnorm + transpose to (C, H, W, T) ----
        ln_transpose_k<<<L, 256, 0, stream>>>(P_X, on_g[s], on_b[s], outp + out_off, L, dim);
        out_off += (size_t)L * dim;
        // ---- patch merging ----
        if (s < 3) {
            const size_t total = (size_t)L * dim;   // = (L/8) * 8*dim
            merge_gather_k<<<(unsigned)((total + 255) / 256), 256, 0, stream>>>(P_X, P_MRG, H, W, T, dim);
            layernorm_k<<<L / 8, 256, 0, stream>>>(P_MRG, down[s].ng, down[s].nb, P_MRG, 8 * dim);
            gemm(P_MRG, 8 * dim, down[s].red_w, 2 * dim, 0, nullptr, nullptr, P_X,
                 L / 8, 2 * dim, 8 * dim, ACT_NONE);
            H /= 2; W /= 2; T /= 2; L /= 8; dim *= 2;
        }
    }
}